// ImageProjectionLayer_42872363549206
// MI455X (gfx1250) — compile-verified
//
#include <hip/hip_runtime.h>
#include <hip/hip_bf16.h>

namespace {

constexpr int H  = 64;
constexpr int W  = 2048;
constexpr int HW = H * W;

constexpr int RADIX_BITS = 6;
constexpr int BINS  = 1 << RADIX_BITS;   // 64
constexpr int BLOCK = 256;               // 8 waves (wave32)
constexpr int IPT   = 16;
constexpr int TILE  = BLOCK * IPT;       // 4096 keys per tile (16 KB)
constexpr int PASSES = 6;                // 6 * 6 = 36 >= 32 bits

constexpr float PI_F    = 3.14159265358979323846f;
constexpr float U_SCALE = (float)((double)(W - 1) / 6.283185307179586476925286766559);
constexpr float VFOV_LO = -0.4328416544740325f;
constexpr float V_SCALE = (float)((double)(H - 1) /
                                  (0.03490658503988659 - (-0.4328416544740325)));

__device__ __forceinline__ int iminD(int a, int b) { return a < b ? a : b; }
__device__ __forceinline__ int imaxD(int a, int b) { return a > b ? a : b; }

// ---------------------------------------------------------------------------
// TDM tile staging (gfx1250 Tensor Data Mover). Wave 0 issues a 1-D
// TENSOR_LOAD_TO_LDS descriptor: tile_dim0 = TILE elems of 4 B, with
// tensor_dim0 = `remaining` so the hardware zero-fills past the end of the
// array (CDNA5 ISA OOB behavior). Guarded by __has_builtin
// with a cooperative-copy fallback so host pass / other toolchains compile.
// This toolchain exposes the 6-arg builtin form:
//   (uint32x4 g0, int32x8 g1, int32x4 g2, int32x4 g3, int32x8 g4, i32 cpol)
// ---------------------------------------------------------------------------
#if defined(__gfx1250__) && __has_builtin(__builtin_amdgcn_tensor_load_to_lds) && \
    __has_builtin(__builtin_amdgcn_s_wait_tensorcnt)
#define IPL_USE_TDM 1
#endif

#ifdef IPL_USE_TDM
typedef __attribute__((ext_vector_type(4))) unsigned uint4v;
typedef __attribute__((ext_vector_type(8))) int      int8v;
typedef __attribute__((ext_vector_type(4))) int      int4v;
typedef __attribute__((address_space(3))) const void lds_cvoid;

__device__ __forceinline__ unsigned ipl_lds_offset(const void* p) {
  // generic -> LDS addrspace cast yields the 32-bit LDS byte offset
  return (unsigned)(unsigned long long)(lds_cvoid*)p;
}

__device__ __forceinline__ void ipl_tdm_issue(const unsigned* __restrict__ gsrc,
                                              unsigned* lds, unsigned remaining) {
  const unsigned long long ga = (unsigned long long)gsrc;
  uint4v g0;
  g0.x = 1u;                                    // count=1 (valid user descriptor)
  g0.y = ipl_lds_offset(lds);                   // lds_addr
  g0.z = (unsigned)(ga & 0xFFFFFFFFull);        // global_addr[31:0]  (bits 95:64)
  g0.w = ((unsigned)(ga >> 32) & 0x01FFFFFFu)   // global_addr[56:32] (bits 120:96)
         | 0x80000000u;                         // type=2 ("image")   (bits 127:126)
  const unsigned d0 = remaining;                // tensor_dim0 -> OOB zero-fill
  const unsigned t0 = (unsigned)TILE;           // tile_dim0
  int8v g1;
  g1[0] = (int)(2u << 16);                      // wg_mask=0, data_size=2 (4 B)
  g1[1] = (int)((d0 & 0xFFFFu) << 16);          // tensor_dim0[15:0] @ bits 63:48
  g1[2] = (int)(((d0 >> 16) & 0xFFFFu)          // tensor_dim0[31:16]
                | (1u << 16));                  // tensor_dim1[15:0] = 1
  g1[3] = (int)((t0 & 0xFFFFu) << 16);          // tensor_dim1[31:16]=0, tile_dim0
  g1[4] = 0;                                    // tile_dim1=0, tile_dim2=0 (1-D)
  g1[5] = (int)t0;                              // tensor_dim0_stride[31:0]
  g1[6] = 0;                                    // strides hi = 0
  g1[7] = 0;
  int4v z4;  z4[0] = 0; z4[1] = 0; z4[2] = 0; z4[3] = 0;
  int8v z8;
  z8[0] = 0; z8[1] = 0; z8[2] = 0; z8[3] = 0;
  z8[4] = 0; z8[5] = 0; z8[6] = 0; z8[7] = 0;
  __builtin_amdgcn_tensor_load_to_lds(g0, g1, z4, z4, z8, 0);
}
#endif

// Stage one 4096-elem u32 tile into LDS; zero-fills past `remaining`.
// Caller-visible after this returns (internal barrier).
__device__ __forceinline__ void ipl_stage_tile(const unsigned* __restrict__ gsrc,
                                               unsigned* lds, int remaining) {
#ifdef IPL_USE_TDM
  if (threadIdx.x < 32u) {                      // wave 0 issues the DMA
    ipl_tdm_issue(gsrc, lds, (unsigned)remaining);
    __builtin_amdgcn_s_wait_tensorcnt(0);
  }
  __syncthreads();
#else
  for (int k = (int)threadIdx.x; k < TILE; k += BLOCK)
    lds[k] = (k < remaining) ? gsrc[k] : 0u;
  __syncthreads();
#endif
}

// Same, for two tiles (keys + vals) with a single TDM wait.
__device__ __forceinline__ void ipl_stage_tile2(const unsigned* __restrict__ gsrcA,
                                                unsigned* ldsA,
                                                const unsigned* __restrict__ gsrcB,
                                                unsigned* ldsB, int remaining) {
#ifdef IPL_USE_TDM
  if (threadIdx.x < 32u) {
    ipl_tdm_issue(gsrcA, ldsA, (unsigned)remaining);
    ipl_tdm_issue(gsrcB, ldsB, (unsigned)remaining);
    __builtin_amdgcn_s_wait_tensorcnt(0);
  }
  __syncthreads();
#else
  for (int k = (int)threadIdx.x; k < TILE; k += BLOCK) {
    ldsA[k] = (k < remaining) ? gsrcA[k] : 0u;
    ldsB[k] = (k < remaining) ? gsrcB[k] : 0u;
  }
  __syncthreads();
#endif
}

// ---------------------------------------------------------------------------
// 1) key/value build: key = IEEE bits of range (range >= 0 -> order-preserving)
// ---------------------------------------------------------------------------
__global__ __launch_bounds__(BLOCK)
void ipl_build_keys(const float* __restrict__ pc,
                    unsigned* __restrict__ keys,
                    unsigned* __restrict__ vals, int n) {
  int i = blockIdx.x * BLOCK + threadIdx.x;
  if (i >= n) return;
  float x = pc[i], y = pc[n + i], z = pc[2 * n + i];
  float r = sqrtf(x * x + y * y + z * z);
  keys[i] = __float_as_uint(r);
  vals[i] = (unsigned)i;
}

// ---------------------------------------------------------------------------
// 2a) per-tile digit histogram -> hist[bin][tile] (keys staged by TDM)
// ---------------------------------------------------------------------------
__global__ __launch_bounds__(BLOCK)
void ipl_radix_hist(const unsigned* __restrict__ keysIn,
                    unsigned* __restrict__ hist,
                    int n, int numTiles, int shift) {
  __shared__ unsigned lh[BINS];
  __shared__ unsigned stage[TILE];
  const int t = threadIdx.x, tile = blockIdx.x;
  const int tileBase = tile * TILE;
  if (t < BINS) lh[t] = 0;
  if (tileBase + TILE < n)                       // warm L2 for a later tile
    __builtin_prefetch(keysIn + tileBase + TILE, 0, 0);
  __syncthreads();
  ipl_stage_tile(keysIn + tileBase, stage, n - tileBase);

  const int base = t * IPT;
  const int cnt = iminD(IPT, imaxD(0, n - (tileBase + base)));
#pragma unroll
  for (int i = 0; i < IPT; ++i)
    if (i < cnt)
      atomicAdd(&lh[(stage[base + i] >> shift) & (BINS - 1)], 1u);
  __syncthreads();
  if (t < BINS) hist[t * numTiles + tile] = lh[t];
}

// ---------------------------------------------------------------------------
// 2b) row scan: hist[bin][*] -> exclusive prefix over tiles; rowTot[bin]=total
// ---------------------------------------------------------------------------
__global__ __launch_bounds__(BLOCK)
void ipl_scan_rows(unsigned* __restrict__ hist,
                   unsigned* __restrict__ rowTot, int numTiles) {
  __shared__ unsigned buf[BLOCK];
  const int b = blockIdx.x, t = threadIdx.x;
  unsigned* row = hist + (size_t)b * numTiles;
  unsigned carry = 0;
  for (int base = 0; base < numTiles; base += BLOCK) {
    const int idx = base + t;
    const unsigned v = (idx < numTiles) ? row[idx] : 0u;
    buf[t] = v;
    __syncthreads();
    for (int off = 1; off < BLOCK; off <<= 1) {   // Hillis-Steele inclusive
      unsigned x = (t >= off) ? buf[t - off] : 0u;
      __syncthreads();
      buf[t] += x;
      __syncthreads();
    }
    if (idx < numTiles) row[idx] = carry + (buf[t] - v);
    const unsigned chunkTot = buf[BLOCK - 1];
    __syncthreads();
    carry += chunkTot;
  }
  if (t == 0) rowTot[b] = carry;
}

// ---------------------------------------------------------------------------
// 2c) exclusive scan of per-bin totals -> binBase
// ---------------------------------------------------------------------------
__global__ __launch_bounds__(BINS)
void ipl_scan_bins(const unsigned* __restrict__ rowTot,
                   unsigned* __restrict__ binBase) {
  __shared__ unsigned buf[BINS];
  const int t = threadIdx.x;
  const unsigned v = rowTot[t];
  buf[t] = v;
  __syncthreads();
  for (int off = 1; off < BINS; off <<= 1) {
    unsigned x = (t >= off) ? buf[t - off] : 0u;
    __syncthreads();
    buf[t] += x;
    __syncthreads();
  }
  binBase[t] = buf[t] - v;   // exclusive
}

// ---------------------------------------------------------------------------
// 2d) stable scatter. Keys+vals tiles staged to LDS by TDM; per-(bin,thread)
//     u16 count table; flattened exclusive scan of [bin][thread] gives exact
//     stable local ranks for the thread-blocked order (= memory order).
// ---------------------------------------------------------------------------
__global__ __launch_bounds__(BLOCK)
void ipl_radix_scatter(const unsigned* __restrict__ keysIn,
                       const unsigned* __restrict__ valsIn,
                       unsigned* __restrict__ keysOut,
                       unsigned* __restrict__ valsOut,
                       const unsigned* __restrict__ histPref,  // [bin][tile] excl
                       const unsigned* __restrict__ binBase,   // [bin]
                       int n, int numTiles, int shift) {
  __shared__ unsigned short F[BINS * BLOCK];   // 32 KB rank table
  __shared__ unsigned stageK[TILE];            // 16 KB keys tile
  __shared__ unsigned stageV[TILE];            // 16 KB vals tile
  __shared__ unsigned partial[BLOCK];
  __shared__ unsigned gOff[BINS];
  __shared__ unsigned binStart[BINS];

  const int t = threadIdx.x, tile = blockIdx.x;
  const int tileBase = tile * TILE;
  if (tileBase + TILE < n)
    __builtin_prefetch(keysIn + tileBase + TILE, 0, 0);

  // stage keys+vals tiles (wait + barrier inside), then all 8 waves zero the rank table
  ipl_stage_tile2(keysIn + tileBase, stageK, valsIn + tileBase, stageV,
                  n - tileBase);
  for (int k = t; k < BINS * BLOCK; k += BLOCK) F[k] = 0;
  __syncthreads();

  const int base = t * IPT;
  const int cnt = iminD(IPT, imaxD(0, n - (tileBase + base)));
  unsigned keys[IPT];
#pragma unroll
  for (int i = 0; i < IPT; ++i) keys[i] = stageK[base + i];

  // count into own column (no atomics: column t owned by thread t)
#pragma unroll
  for (int i = 0; i < IPT; ++i)
    if (i < cnt) {
      const unsigned d = (keys[i] >> shift) & (BINS - 1);
      F[d * BLOCK + t] = (unsigned short)(F[d * BLOCK + t] + 1);
    }
  __syncthreads();

  // flattened exclusive scan of F (BINS*BLOCK entries), SEG entries per thread
  constexpr int SEG = (BINS * BLOCK) / BLOCK;   // 64
  unsigned s = 0;
  for (int k = 0; k < SEG; ++k) s += F[t * SEG + k];
  partial[t] = s;
  __syncthreads();
  for (int off = 1; off < BLOCK; off <<= 1) {
    unsigned x = (t >= off) ? partial[t - off] : 0u;
    __syncthreads();
    partial[t] += x;
    __syncthreads();
  }
  unsigned run = partial[t] - s;   // exclusive prefix of segment sums
  for (int k = 0; k < SEG; ++k) {
    const unsigned v = F[t * SEG + k];
    F[t * SEG + k] = (unsigned short)run;
    run += v;
  }
  __syncthreads();

  // snapshot per-bin tile-start BEFORE emit (F[d*BLOCK+0] doubles as thread
  // 0's running counter during emit), and fetch global bin/tile offsets
  if (t < BINS) {
    binStart[t] = F[t * BLOCK];
    gOff[t]     = binBase[t] + histPref[(size_t)t * numTiles + tile];
  }
  __syncthreads();

  // emit in order; per-column counters give stable ranks
#pragma unroll
  for (int i = 0; i < IPT; ++i)
    if (i < cnt) {
      const unsigned d   = (keys[i] >> shift) & (BINS - 1);
      const int      idx = (int)d * BLOCK + t;
      const unsigned r   = F[idx];
      F[idx] = (unsigned short)(r + 1);
      const unsigned dest = gOff[d] + (r - binStart[d]);
      keysOut[dest] = keys[i];
      valsOut[dest] = stageV[base + i];
    }
}

// ---------------------------------------------------------------------------
// 3) winner init + projection (u,v outputs + per-pixel min-rank z-buffer)
// ---------------------------------------------------------------------------
__global__ __launch_bounds__(BLOCK)
void ipl_init_winner(int* __restrict__ winner, int n) {
  int p = blockIdx.x * BLOCK + threadIdx.x;
  if (p < HW) winner[p] = n;
}

__global__ __launch_bounds__(BLOCK)
void ipl_project(const unsigned* __restrict__ keys,
                 const unsigned* __restrict__ vals,
                 const float* __restrict__ pc,
                 float* __restrict__ u_out, float* __restrict__ v_out,
                 int* __restrict__ winner, int n) {
  const int i = blockIdx.x * BLOCK + threadIdx.x;
  if (i >= n) return;
  const unsigned j = vals[i];
  const float x = pc[j], y = pc[n + j], z = pc[2 * n + j];
  const float u  = (atan2f(y, x) + PI_F) * U_SCALE;
  const float xy = sqrtf(x * x + y * y);
  const float v  = (atan2f(z, xy) - VFOV_LO) * V_SCALE;
  u_out[i] = u;
  v_out[i] = v;
  const float ur = rintf(u);   // round-half-even == jnp.round
  const float vr = rintf(v);
  if (ur >= 0.0f && ur <= (float)(W - 1) && vr >= 0.0f && vr <= (float)(H - 1)) {
    const int pix = (int)(vr * (float)W + ur);
    atomicMin(&winner[pix], i);   // min: order-independent -> deterministic
  }
}

// ---------------------------------------------------------------------------
// 4) gather winners into (1, 5, H, W) image
// ---------------------------------------------------------------------------
__global__ __launch_bounds__(BLOCK)
void ipl_image(const unsigned* __restrict__ keys,
               const unsigned* __restrict__ vals,
               const float* __restrict__ pc,
               const int* __restrict__ winner,
               float* __restrict__ img, int n) {
  const int p = blockIdx.x * BLOCK + threadIdx.x;
  if (p >= HW) return;
  const int w = winner[p];
  float c0 = 0.f, c1 = 0.f, c2 = 0.f, c3 = 0.f, c4 = 0.f;
  if (w < n) {
    const unsigned j = vals[w];
    c0 = pc[j];
    c1 = pc[n + j];
    c2 = pc[2 * n + j];
    c3 = pc[3 * n + j];
    c4 = __uint_as_float(keys[w]);   // range
  }
  img[0 * HW + p] = c0;
  img[1 * HW + p] = c1;
  img[2 * HW + p] = c2;
  img[3 * HW + p] = c3;
  img[4 * HW + p] = c4;
}

} // namespace

extern "C" void kernel_launch(void* const* d_in, const int* in_sizes, int n_in,
                              void* d_out, int out_size, void* d_ws, size_t ws_size,
                              hipStream_t stream) {
  (void)n_in; (void)out_size; (void)ws_size;
  const float* pc = (const float*)d_in[0];
  const int n = in_sizes[0] / 4;                 // (1,4,N) -> N
  const int numTiles = (n + TILE - 1) / TILE;

  // ---- workspace partition (256B aligned) ----
  char* wsb = (char*)d_ws;
  size_t off = 0;
  auto take = [&](size_t bytes) -> void* {
    void* p = wsb + off;
    off += (bytes + 255) & ~(size_t)255;
    return p;
  };
  unsigned* keysA  = (unsigned*)take((size_t)n * 4);
  unsigned* keysB  = (unsigned*)take((size_t)n * 4);
  unsigned* valsA  = (unsigned*)take((size_t)n * 4);
  unsigned* valsB  = (unsigned*)take((size_t)n * 4);
  unsigned* hist   = (unsigned*)take((size_t)BINS * numTiles * 4);
  unsigned* rowTot = (unsigned*)take((size_t)BINS * 4);
  unsigned* binB   = (unsigned*)take((size_t)BINS * 4);
  int*      winner = (int*)take((size_t)HW * 4);

  float* img   = (float*)d_out;                  // 5*H*W
  float* u_out = img + (size_t)5 * HW;           // N
  float* v_out = u_out + (size_t)n;              // N

  const int gPts = (n + BLOCK - 1) / BLOCK;
  const int gPix = (HW + BLOCK - 1) / BLOCK;

  ipl_build_keys<<<gPts, BLOCK, 0, stream>>>(pc, keysA, valsA, n);

  unsigned *kCur = keysA, *vCur = valsA, *kAlt = keysB, *vAlt = valsB;
  for (int p = 0; p < PASSES; ++p) {
    const int shift = p * RADIX_BITS;
    ipl_radix_hist<<<numTiles, BLOCK, 0, stream>>>(kCur, hist, n, numTiles, shift);
    ipl_scan_rows<<<BINS, BLOCK, 0, stream>>>(hist, rowTot, numTiles);
    ipl_scan_bins<<<1, BINS, 0, stream>>>(rowTot, binB);
    ipl_radix_scatter<<<numTiles, BLOCK, 0, stream>>>(kCur, vCur, kAlt, vAlt,
                                                      hist, binB, n, numTiles, shift);
    unsigned* tk = kCur; kCur = kAlt; kAlt = tk;
    unsigned* tv = vCur; vCur = vAlt; vAlt = tv;
  }
  // PASSES is even -> sorted pairs are back in (keysA, valsA) == (kCur, vCur)

  ipl_init_winner<<<gPix, BLOCK, 0, stream>>>(winner, n);
  ipl_project<<<gPts, BLOCK, 0, stream>>>(kCur, vCur, pc, u_out, v_out, winner, n);
  ipl_image<<<gPix, BLOCK, 0, stream>>>(kCur, vCur, pc, winner, img, n);
}